// RCNNLoss_75222057222814
// MI455X (gfx1250) — compile-verified
//
#include <hip/hip_runtime.h>
#include <hip/hip_bf16.h>

#define NCLS 21      // classes (VOC 20 + background)
#define TILE 256     // rows staged per block iteration
#define NBLK 256     // grid size of the partial kernel

typedef __attribute__((ext_vector_type(2))) float v2f;
typedef __attribute__((ext_vector_type(8))) float v8f;
typedef __attribute__((ext_vector_type(4))) int   v4i;

// Address-space-qualified pointee types matching the async builtin's params
// (diagnostic showed: pointer to int __vector(4) in the device address space).
typedef __attribute__((address_space(1))) v4i as1_v4i;   // global
typedef __attribute__((address_space(3))) v4i as3_v4i;   // LDS

#if __has_builtin(__builtin_amdgcn_global_load_async_to_lds_b128)
#define HAVE_ASYNC_LDS 1
#else
#define HAVE_ASYNC_LDS 0
#endif

__device__ __forceinline__ void async_copy_b128(const float4* gsrc, float4* ldst) {
#if HAVE_ASYNC_LDS
    __builtin_amdgcn_global_load_async_to_lds_b128(
        (as1_v4i*)gsrc, (as3_v4i*)ldst, /*offset=*/0, /*cpol=*/0);
#else
    *ldst = *gsrc;
#endif
}

__device__ __forceinline__ void async_wait_all() {
#if HAVE_ASYNC_LDS
#if __has_builtin(__builtin_amdgcn_s_wait_asynccnt)
    __builtin_amdgcn_s_wait_asynccnt(0);
#else
    asm volatile("s_wait_asynccnt 0" ::: "memory");
#endif
#endif
}

__device__ __forceinline__ float smooth_l1(float x) {
    float a = fabsf(x);
    return a < 1.0f ? 0.5f * x * x : a - 0.5f;
}

// Wave32 all-lane sum via the matrix pipe:
//   D = ones(16x4) * B(4x16) + 0
// Every lane contributes one value in B; D[m][n] = column-sum(n) gathers all
// 32 values into 16 column sums (replicated across D rows), then 4 xor
// shuffles fold the 16 columns. EXEC must be all 1s at the call site.
__device__ __forceinline__ float wave_sum_wmma(float v) {
    v2f a; a[0] = 1.0f; a[1] = 1.0f;   // A = ones 16x4 (2 VGPRs/lane)
    v2f b; b[0] = v;    b[1] = 0.0f;   // B = 4x16, one live value per lane
    v8f c = {};
    c = __builtin_amdgcn_wmma_f32_16x16x4_f32(
            /*neg_a=*/false, a, /*neg_b=*/false, b,
            /*c_mod=*/(short)0, c, /*reuse_a=*/false, /*reuse_b=*/false);
    float s = c[0];                    // lane n (mod 16): column-sum(n)
    s += __shfl_xor(s, 1, 32);
    s += __shfl_xor(s, 2, 32);
    s += __shfl_xor(s, 4, 32);
    s += __shfl_xor(s, 8, 32);
    return s;                          // total in every lane
}

__global__ __launch_bounds__(256) void rcnn_loss_partial(
    const float* __restrict__ logits,   // [N, 21]
    const float* __restrict__ pboxes,   // [N, 21, 4]
    const float* __restrict__ tboxes,   // [N, 5]
    float* __restrict__ partials,       // [NBLK, 3] = {ce, loc, cnt}
    int n) {
    __shared__ __align__(16) float s_logits[TILE * NCLS];  // 21504 B
    __shared__ __align__(16) float s_tb[TILE * 5];         //  5120 B
    __shared__ float s_red[8][3];

    const int tid = threadIdx.x;
    const int numTiles = n / TILE;     // N is a multiple of 256

    float ce_acc = 0.0f, loc_acc = 0.0f, cnt_acc = 0.0f;

    for (int tile = blockIdx.x; tile < numTiles; tile += gridDim.x) {
        const int row0 = tile * TILE;

        // Stage logits tile: 256*21 floats = 1344 x b128, 16B-aligned.
        // global_load_async_to_lds_b128 (ASYNCcnt): no VGPR round-trip,
        // LDS written directly per lane.
        const float4* gsrc = (const float4*)(logits + (size_t)row0 * NCLS);
        float4* ldst = (float4*)s_logits;
        #pragma unroll 2
        for (int i = tid; i < (TILE * NCLS) / 4; i += 256)
            async_copy_b128(gsrc + i, ldst + i);

        // Stage true_boxes tile: 256*5 floats = 320 x b128, 16B-aligned.
        const float4* gtb = (const float4*)(tboxes + (size_t)row0 * 5);
        float4* ltb = (float4*)s_tb;
        for (int i = tid; i < (TILE * 5) / 4; i += 256)
            async_copy_b128(gtb + i, ltb + i);

        // Hint the next tile of logits into cache (global_prefetch_b8).
        const int nextTile = tile + gridDim.x;
        if (nextTile < numTiles)
            __builtin_prefetch(logits + (size_t)nextTile * TILE * NCLS + tid * NCLS, 0, 0);

        async_wait_all();      // s_wait_asynccnt 0: our LDS writes landed
        __syncthreads();       // everyone's writes visible

        // Per-row compute. LDS read strides 21 and 5 are coprime with 64
        // banks -> conflict-free scalar ds loads.
        const float* lg = &s_logits[tid * NCLS];
        const float* tb = &s_tb[tid * 5];
        const int label = (int)tb[4];

        float m = lg[0];
        #pragma unroll
        for (int c1 = 1; c1 < NCLS; ++c1) m = fmaxf(m, lg[c1]);
        float se = 0.0f;
        #pragma unroll
        for (int c1 = 0; c1 < NCLS; ++c1) se += __expf(lg[c1] - m);
        ce_acc += (m + __logf(se)) - lg[label];   // -log p(label)

        // Gather the true-class box: one aligned 16B load out of the 336B row
        // (key data-movement saving: ~33MB touched vs 88MB streamed).
        const size_t row = (size_t)row0 + (size_t)tid;
        const float4 pb = ((const float4*)pboxes)[row * NCLS + (size_t)label];
        const float rl = smooth_l1(tb[0] - pb.x) + smooth_l1(tb[1] - pb.y) +
                         smooth_l1(tb[2] - pb.z) + smooth_l1(tb[3] - pb.w);
        const float msk = (label != 0) ? 1.0f : 0.0f;
        loc_acc += rl * msk;
        cnt_acc += msk;

        __syncthreads();   // LDS reused next iteration
    }

    // Block reduction: WMMA per wave (EXEC all-1s here), then LDS combine.
    const float wce  = wave_sum_wmma(ce_acc);
    const float wloc = wave_sum_wmma(loc_acc);
    const float wcnt = wave_sum_wmma(cnt_acc);
    const int wave = tid >> 5, lane = tid & 31;
    if (lane == 0) {
        s_red[wave][0] = wce; s_red[wave][1] = wloc; s_red[wave][2] = wcnt;
    }
    __syncthreads();
    if (tid == 0) {
        float a0 = 0.0f, a1 = 0.0f, a2 = 0.0f;
        for (int w = 0; w < 8; ++w) {
            a0 += s_red[w][0]; a1 += s_red[w][1]; a2 += s_red[w][2];
        }
        partials[3 * blockIdx.x + 0] = a0;
        partials[3 * blockIdx.x + 1] = a1;
        partials[3 * blockIdx.x + 2] = a2;
    }
}

__global__ __launch_bounds__(256) void rcnn_loss_finalize(
    const float* __restrict__ partials, int nPart,
    float* __restrict__ out, float invN) {
    __shared__ float s_red[8][3];
    const int tid = threadIdx.x;

    float a0 = 0.0f, a1 = 0.0f, a2 = 0.0f;
    for (int i = tid; i < nPart; i += 256) {
        a0 += partials[3 * i + 0];
        a1 += partials[3 * i + 1];
        a2 += partials[3 * i + 2];
    }
    const float w0 = wave_sum_wmma(a0);
    const float w1 = wave_sum_wmma(a1);
    const float w2 = wave_sum_wmma(a2);
    const int wave = tid >> 5, lane = tid & 31;
    if (lane == 0) {
        s_red[wave][0] = w0; s_red[wave][1] = w1; s_red[wave][2] = w2;
    }
    __syncthreads();
    if (tid == 0) {
        float ce = 0.0f, loc = 0.0f, cnt = 0.0f;
        for (int w = 0; w < 8; ++w) {
            ce += s_red[w][0]; loc += s_red[w][1]; cnt += s_red[w][2];
        }
        out[0] = ce * invN + loc / fmaxf(cnt, 1.0f);
    }
}

extern "C" void kernel_launch(void* const* d_in, const int* in_sizes, int n_in,
                              void* d_out, int out_size, void* d_ws, size_t ws_size,
                              hipStream_t stream) {
    (void)n_in; (void)out_size; (void)ws_size;
    const float* logits = (const float*)d_in[0];   // [N,21]
    const float* pboxes = (const float*)d_in[1];   // [N,21,4]
    const float* tboxes = (const float*)d_in[2];   // [N,5]
    const int n = in_sizes[2] / 5;                 // N from true_boxes

    float* partials = (float*)d_ws;                // NBLK*3 floats of scratch

    rcnn_loss_partial<<<NBLK, 256, 0, stream>>>(logits, pboxes, tboxes, partials, n);
    rcnn_loss_finalize<<<1, 256, 0, stream>>>(partials, NBLK, (float*)d_out,
                                              1.0f / (float)n);
}